// SimpleMPNN_30081950941813
// MI455X (gfx1250) — compile-verified
//
#include <hip/hip_runtime.h>
#include <hip/hip_bf16.h>

typedef float v2f __attribute__((ext_vector_type(2)));
typedef float v8f __attribute__((ext_vector_type(8)));

#define ALPHA_C 0.5f
#define EPS_C 1e-12f

// ---------------------------------------------------------------------------
// Kernel 1: q_hat = q / max(||q||, eps).  One block, D<=128 threads.
// ---------------------------------------------------------------------------
__global__ void qnorm_kernel(const float* __restrict__ q,
                             float* __restrict__ qhat, int D) {
    __shared__ float red[128];
    int t = threadIdx.x;
    float v = (t < D) ? q[t] : 0.0f;
    red[t] = v * v;
    __syncthreads();
    for (int s = 64; s > 0; s >>= 1) {
        if (t < s) red[t] += red[t + s];
        __syncthreads();
    }
    float n = fmaxf(sqrtf(red[0]), EPS_C);
    if (t < D) qhat[t] = v / n;
}

// ---------------------------------------------------------------------------
// Kernel 2: edge similarities via V_WMMA_F32_16X16X4_F32 + dst-degree count.
//
// One wave handles 16 edges. B (4x16, KxN) holds the edge rows: lane n<16
// loads ea[n][k0], ea[n][k0+1]; lane n+16 loads ea[n][k0+2], ea[n][k0+3]
// (coalesced 8B loads). A (16x4) broadcasts q_hat chunk. After 32 K-steps
// every lane holds dot(q_hat, ea[lane&15]) in C[0] — no shuffle reduction.
// ||ea||^2 is accumulated per-lane and combined with one __shfl_xor(16).
// ---------------------------------------------------------------------------
__global__ void edge_sim_kernel(const float* __restrict__ edge_attr,
                                const float* __restrict__ qhat,
                                const int* __restrict__ dst,
                                float* __restrict__ sim,
                                float* __restrict__ deg,
                                int E, int D) {
    __shared__ float qs[128];
    int t = threadIdx.x;
    if (t < D) qs[t] = qhat[t];
    __syncthreads();

    int lane  = t & 31;
    int wave  = t >> 5;
    int group = blockIdx.x * 8 + wave;
    int numGroups = (E + 15) >> 4;
    if (group >= numGroups) return;           // wave-uniform: EXEC stays all-1s

    int  base = group << 4;
    int  row  = lane & 15;                    // edge within group
    int  koff = (lane >> 4) << 1;             // 0 for lanes 0-15, 2 for 16-31
    long eIdx = base + row;
    if (eIdx > (long)E - 1) eIdx = (long)E - 1;   // clamp (reads only)
    const float* ea = edge_attr + eIdx * (long)D;

    v8f c = {0.f, 0.f, 0.f, 0.f, 0.f, 0.f, 0.f, 0.f};
    float n2 = 0.0f;
    for (int k0 = 0; k0 < D; k0 += 4) {
        float bx = ea[k0 + koff];
        float by = ea[k0 + koff + 1];
        v2f b; b[0] = bx;              b[1] = by;
        v2f a; a[0] = qs[k0 + koff];   a[1] = qs[k0 + koff + 1];
        c = __builtin_amdgcn_wmma_f32_16x16x4_f32(
                /*neg_a=*/false, a, /*neg_b=*/false, b,
                /*c_mod=*/(short)0, c, /*reuse_a=*/false, /*reuse_b=*/false);
        n2 = fmaf(bx, bx, n2);
        n2 = fmaf(by, by, n2);
    }
    n2 += __shfl_xor(n2, 16, 32);             // full ||ea||^2 in every lane

    float s = c[0] / fmaxf(sqrtf(n2), EPS_C); // q already normalized

    int e = base + lane;
    if (lane < 16 && e < E) {
        sim[e] = s;
        atomicAdd(&deg[dst[e]], 1.0f);
    }
}

// ---------------------------------------------------------------------------
// Kernel 3: scatter-add  acc[dst] += x[src] * sim.  One wave per edge:
// 32 lanes x float4 = one 512B node row, coalesced; 4 f32 atomics per lane
// land in the L2-resident accumulator. Prefetch next gather row.
// ---------------------------------------------------------------------------
__global__ void scatter_kernel(const float* __restrict__ x,
                               const float* __restrict__ sim,
                               const int* __restrict__ src,
                               const int* __restrict__ dst,
                               float* __restrict__ acc,
                               int E, int D) {
    int lane = threadIdx.x & 31;
    int wavesPerBlock = blockDim.x >> 5;
    int wave = blockIdx.x * wavesPerBlock + (threadIdx.x >> 5);
    int nw   = gridDim.x * wavesPerBlock;
    int nquads = D >> 2;

    for (int e = wave; e < E; e += nw) {
        int en = e + nw;
        if (lane == 0 && en < E) {
            __builtin_prefetch(x + (long)src[en] * D, 0, 0);  // global_prefetch_b8
        }
        int   s = src[e];
        int   d = dst[e];
        float w = sim[e];
        const float4* xr = (const float4*)(x + (long)s * D);
        for (int cq = lane; cq < nquads; cq += 32) {
            float4 v = xr[cq];
            float* ar = acc + (long)d * D + (long)cq * 4;
            atomicAdd(ar + 0, v.x * w);
            atomicAdd(ar + 1, v.y * w);
            atomicAdd(ar + 2, v.z * w);
            atomicAdd(ar + 3, v.w * w);
        }
    }
}

// ---------------------------------------------------------------------------
// Kernel 4: x_out = ALPHA*x_old + (1-ALPHA)*acc/max(deg,1).  float4 stream.
// (xold/xout may alias in-place per element — no __restrict__ on them.)
// ---------------------------------------------------------------------------
__global__ void blend_kernel(const float* xold,
                             const float* __restrict__ acc,
                             const float* __restrict__ deg,
                             float* xout, int N, int D) {
    long i = (long)blockIdx.x * blockDim.x + threadIdx.x;   // float4 index
    long total = (long)N * D >> 2;
    if (i >= total) return;
    int qpr  = D >> 2;
    int node = (int)(i / qpr);
    float inv = (1.0f - ALPHA_C) / fmaxf(deg[node], 1.0f);
    float4 a = ((const float4*)xold)[i];
    float4 b = ((const float4*)acc)[i];
    float4 r;
    r.x = ALPHA_C * a.x + b.x * inv;
    r.y = ALPHA_C * a.y + b.y * inv;
    r.z = ALPHA_C * a.z + b.z * inv;
    r.w = ALPHA_C * a.w + b.w * inv;
    ((float4*)xout)[i] = r;
}

// ---------------------------------------------------------------------------
extern "C" void kernel_launch(void* const* d_in, const int* in_sizes, int n_in,
                              void* d_out, int out_size, void* d_ws, size_t ws_size,
                              hipStream_t stream) {
    (void)n_in; (void)out_size; (void)ws_size;
    const float* x         = (const float*)d_in[0];
    const float* edge_attr = (const float*)d_in[1];
    const float* q         = (const float*)d_in[2];
    const int*   ei        = (const int*)d_in[3];

    int D = in_sizes[2];
    int E = in_sizes[3] / 2;
    int N = in_sizes[0] / D;
    const int* src = ei;
    const int* dst = ei + E;

    // workspace layout (floats, 256-float aligned blocks)
    float* ws   = (float*)d_ws;
    size_t off  = 0;
    float* qhat = ws + off; off += 256;
    float* deg  = ws + off; off += ((size_t)N + 255) / 256 * 256;
    float* sim  = ws + off; off += ((size_t)E + 255) / 256 * 256;
    float* acc  = ws + off;
    float* out  = (float*)d_out;

    (void)hipMemsetAsync(deg, 0, (size_t)N * sizeof(float), stream);
    (void)hipMemsetAsync(acc, 0, (size_t)N * D * sizeof(float), stream);

    qnorm_kernel<<<1, 128, 0, stream>>>(q, qhat, D);

    int numGroups = (E + 15) / 16;
    int simBlocks = (numGroups + 7) / 8;
    edge_sim_kernel<<<simBlocks, 256, 0, stream>>>(edge_attr, qhat, dst,
                                                   sim, deg, E, D);

    int scatterBlocks = 2048;
    long total4 = (long)N * D / 4;
    int blendBlocks = (int)((total4 + 255) / 256);

    // ---- layer 1: x0 -> out
    scatter_kernel<<<scatterBlocks, 256, 0, stream>>>(x, sim, src, dst, acc, E, D);
    blend_kernel<<<blendBlocks, 256, 0, stream>>>(x, acc, deg, out, N, D);

    // ---- layer 2: out -> out (in-place blend, per-element)
    (void)hipMemsetAsync(acc, 0, (size_t)N * D * sizeof(float), stream);
    scatter_kernel<<<scatterBlocks, 256, 0, stream>>>(out, sim, src, dst, acc, E, D);
    blend_kernel<<<blendBlocks, 256, 0, stream>>>(out, acc, deg, out, N, D);
}